// PairwiseContactPredictionHead_61125974557639
// MI455X (gfx1250) — compile-verified
//
#include <hip/hip_runtime.h>
#include <hip/hip_bf16.h>

// out[i,j,o] = sum_h x[i,h]*W1[o,h]*x[j,h] + b[o], i,j in [1,510]
// (diff term cancels under (pred+pred^T)/2 symmetrization)
// Two f32 GEMMs via V_WMMA_F32_16X16X4_F32, one wave per 16x16 tile.

typedef float v2f __attribute__((ext_vector_type(2)));
typedef float v8f __attribute__((ext_vector_type(8)));

#define N_SEQ 512
#define H_DIM 768
#define OUT_N 510

__global__ __launch_bounds__(128)
void pairwise_contact_wmma(const float* __restrict__ x,
                           const float* __restrict__ W,
                           const float* __restrict__ bias,
                           float* __restrict__ out) {
    const int lane = threadIdx.x & 31;
    const int wave = threadIdx.x >> 5;
    const int l = lane & 15;   // row/col index within tile
    const int h = lane >> 4;   // K half selector

    const int tile_j = blockIdx.x * 4 + wave;  // 0..31
    const int tile_i = blockIdx.y;             // 0..31
    const int i_base = tile_i * 16;
    const int j_base = tile_j * 16;

    const float* __restrict__ arow = x + (size_t)(i_base + l) * H_DIM;
    const float* __restrict__ brow = x + (size_t)(j_base + l) * H_DIM;
    const float* __restrict__ w1o0 = W;              // W[0, 0:768]
    const float* __restrict__ w1o1 = W + 2 * H_DIM;  // W[1, 0:768] (row stride 1536)

    v8f acc0 = {};
    v8f acc1 = {};

    #pragma unroll 4
    for (int k0 = 0; k0 < H_DIM; k0 += 4) {
        const int kk = k0 + 2 * h;
        // f32 16x16x4 layout: lane(l,h), VGPR v holds K = 2h+v  -> contiguous float2
        v2f a  = *reinterpret_cast<const v2f*>(arow + kk);
        v2f bb = *reinterpret_cast<const v2f*>(brow + kk);
        v2f w0 = *reinterpret_cast<const v2f*>(w1o0 + kk);
        v2f w1 = *reinterpret_cast<const v2f*>(w1o1 + kk);
        v2f a0 = a * w0;  // A_0 = X .* W1[0]
        v2f a1 = a * w1;  // A_1 = X .* W1[1]
        // 8 args: (neg_a, A, neg_b, B, c_mod, C, reuse_a, reuse_b)
        acc0 = __builtin_amdgcn_wmma_f32_16x16x4_f32(false, a0, false, bb,
                                                     (short)0, acc0, false, false);
        acc1 = __builtin_amdgcn_wmma_f32_16x16x4_f32(false, a1, false, bb,
                                                     (short)0, acc1, false, false);
    }

    const float b0 = bias[0];
    const float b1 = bias[1];

    const int j = j_base + l;
    if (j >= 1 && j <= OUT_N) {
        #pragma unroll
        for (int r = 0; r < 8; ++r) {
            const int i = i_base + r + 8 * h;  // C/D layout: VGPR r -> M = r + 8*h
            if (i >= 1 && i <= OUT_N) {
                v2f v;
                v.x = acc0[r] + b0;  // o = 0
                v.y = acc1[r] + b1;  // o = 1
                *reinterpret_cast<v2f*>(out + ((size_t)(i - 1) * OUT_N + (j - 1)) * 2) = v;
            }
        }
    }
}

extern "C" void kernel_launch(void* const* d_in, const int* in_sizes, int n_in,
                              void* d_out, int out_size, void* d_ws, size_t ws_size,
                              hipStream_t stream) {
    const float* x    = (const float*)d_in[0];  // (1, 512, 768) f32
    const float* W    = (const float*)d_in[1];  // (2, 1536) f32
    const float* bias = (const float*)d_in[2];  // (2,) f32
    float* out = (float*)d_out;                 // (1, 510, 510, 2) f32

    dim3 grid(8, 32, 1);   // grid.x*4 waves cover 32 j-tiles; grid.y = 32 i-tiles
    dim3 block(128, 1, 1); // 4 waves, each owns one 16x16 tile (both o outputs)
    pairwise_contact_wmma<<<grid, block, 0, stream>>>(x, W, bias, out);
}